// DiffusionInteractionBlock_31233002176562
// MI455X (gfx1250) — compile-verified
//
#include <hip/hip_runtime.h>
#include <hip/hip_bf16.h>

typedef __attribute__((ext_vector_type(2))) float v2f;
typedef __attribute__((ext_vector_type(8))) float v8f;

#define XSTR 168   // padded row stride for the 16x164 mlp_in tile
#define HSTR 68    // padded row stride for the 16x64 hidden tile
#define SSTR 68    // padded row stride for s_up tile
#define VSTR 196   // padded row stride for v_up tile
#define WAVE_FLOATS (16*XSTR + 16*HSTR + 16*SSTR + 16*VSTR + 64 + 32)

__device__ __forceinline__ v8f wmma4(v2f a, v2f b, v8f c) {
  // V_WMMA_F32_16X16X4_F32 : D = A(16x4) x B(4x16) + C, fp32 throughout
  return __builtin_amdgcn_wmma_f32_16x16x4_f32(false, a, false, b, (short)0, c, false, false);
}

__device__ __forceinline__ float silu_f(float x) {
  return x / (1.0f + __expf(-x));
}

// One 16x16 output tile: A (16xK) staged in LDS (row stride XS, ds_load_b64),
// B from pair-packed weights Wp[k/2][col][2] (one global_load_b64 per step).
template<int K, int LD, int XS>
__device__ __forceinline__ v8f mlp_tile(const float* Xb, const float* __restrict__ Wp,
                                        int col, int lo, int hi) {
  v8f acc = {};
  #pragma unroll 4
  for (int k = 0; k < K; k += 4) {
    v2f a = *(const v2f*)&Xb[lo*XS + k + 2*hi];
    v2f b = *(const v2f*)&Wp[(((size_t)(k >> 1) + hi)*LD + col)*2];
    acc = wmma4(a, b, acc);
  }
  return acc;
}

// ---------------------------------------------------------------------------
// Kernel 0: pack W (rows x cols, row-major) into K-pair-interleaved layout
//   Wp[(k/2)*cols*2 + col*2 + (k&1)] = W[k][col]   (zero-padded past `rows`)
// launched over padded_rows * cols threads
// ---------------------------------------------------------------------------
__global__ __launch_bounds__(256) void pack_pairs(const float* __restrict__ W,
                                                  float* __restrict__ Wp,
                                                  int rows, int cols, int total) {
  int t = blockIdx.x * blockDim.x + threadIdx.x;
  if (t >= total) return;
  int k = t / cols;
  int col = t - k * cols;
  float v = (k < rows) ? W[(size_t)k * cols + col] : 0.0f;
  Wp[(((size_t)(k >> 1)) * cols + col) * 2 + (k & 1)] = v;
}

// ---------------------------------------------------------------------------
// Kernel 1: node precompute
//   node_scalars = s @ Wsc ; s_up = s @ Wup_s ; v_up[:,g,c] = sum_f v[:,f,c] Wup_v[f,g]
// ---------------------------------------------------------------------------
__global__ __launch_bounds__(256) void node_precompute(
    const float* __restrict__ nf, const float* __restrict__ Wsc,
    const float* __restrict__ Wus, const float* __restrict__ Wuv,
    float* __restrict__ ns, float* __restrict__ su, float* __restrict__ vu)
{
  int t = blockIdx.x * blockDim.x + threadIdx.x;
  int n = t >> 6;
  int g = t & 63;
  const float* row = nf + (size_t)n * 256;
  float asc = 0.f, asu = 0.f, av0 = 0.f, av1 = 0.f, av2 = 0.f;
  for (int f = 0; f < 64; ++f) {
    float s = row[f];
    asc += s * Wsc[f*64 + g];
    asu += s * Wus[f*64 + g];
    float w = Wuv[f*64 + g];
    av0 += row[64 + f*3 + 0] * w;
    av1 += row[64 + f*3 + 1] * w;
    av2 += row[64 + f*3 + 2] * w;
  }
  ns[(size_t)n*64 + g] = asc;
  su[(size_t)n*64 + g] = asu;
  vu[(size_t)n*192 + g*3 + 0] = av0;
  vu[(size_t)n*192 + g*3 + 1] = av1;
  vu[(size_t)n*192 + g*3 + 2] = av2;
}

// ---------------------------------------------------------------------------
// Kernel 2: per-edge MLP (fp32 WMMA) + tensor product + atomic scatter
// block = 64 threads = 2 waves; each wave owns a 16-edge tile and its own
// LDS staging region (wave-private -> no block barriers; LDS in-order per wave)
// ---------------------------------------------------------------------------
__global__ __launch_bounds__(64) void edge_mlp_tp(
    const float* __restrict__ ns, const float* __restrict__ su, const float* __restrict__ vu,
    const float* __restrict__ ea, const float* __restrict__ ef, const float* __restrict__ len,
    const int*   __restrict__ ei,
    const float* __restrict__ W1p, const float* __restrict__ b1,
    const float* __restrict__ W2p, const float* __restrict__ b2,
    const float* __restrict__ W3p,
    float* __restrict__ M0, float* __restrict__ M1, int E)
{
  __shared__ float smem[2 * WAVE_FLOATS];
  const int lane = threadIdx.x & 31;
  const int wave = threadIdx.x >> 5;
  const int lo = lane & 15;     // WMMA: A row M / B,C,D column N
  const int hi = lane >> 4;     // WMMA: K-half selector

  float* X  = smem + wave * WAVE_FLOATS;   // 16 x 164 mlp_in tile (later H2)
  float* H  = X  + 16 * XSTR;              // 16 x 64 hidden tile (H1)
  float* SS = H  + 16 * HSTR;              // 16 x 64 s_up[snd]
  float* VS = SS + 16 * SSTR;              // 16 x 192 v_up[snd] (f*3+c)
  float* AT = VS + 16 * VSTR;              // 16 x 4 edge_attrs
  int* SND  = (int*)(AT + 64);
  int* RCV  = SND + 16;

  const int eBase = (blockIdx.x * 2 + wave) * 16;

  // pull weights toward L2 early (global_prefetch_b8)
  __builtin_prefetch(W1p + (size_t)threadIdx.x * 64, 0, 0);
  __builtin_prefetch(W3p + (size_t)threadIdx.x * 64, 0, 0);

  if (lane < 16) {
    int e = eBase + lane;
    SND[lane] = ei[e];
    RCV[lane] = ei[E + e];
    AT[lane*4+0] = ea[(size_t)e*4 + 0];
    AT[lane*4+1] = ea[(size_t)e*4 + 1];
    AT[lane*4+2] = ea[(size_t)e*4 + 2];
    AT[lane*4+3] = ea[(size_t)e*4 + 3];
  }

  // build mlp_in: [ns[snd](64) | ns[rcv](64) | edge_feats(32) | length(1) | pad->164]
  {
    int row = lane >> 1;
    int c0 = (lane & 1) * 82;
    int e = eBase + row;
    int snd = SND[row];
    int rcv = RCV[row];
    float* xr = X + row * XSTR;
    for (int i = 0; i < 82; ++i) {
      int c = c0 + i;
      float v;
      if (c < 64)        v = ns[(size_t)snd*64 + c];
      else if (c < 128)  v = ns[(size_t)rcv*64 + (c - 64)];
      else if (c < 160)  v = ef[(size_t)e*32 + (c - 128)];
      else if (c == 160) v = len[e];
      else               v = 0.0f;
      xr[c] = v;
    }
  }
  // stage sender s_up / v_up for the tensor product
  for (int idx = lane; idx < 16*64; idx += 32) {
    int m = idx >> 6, c = idx & 63;
    SS[m*SSTR + c] = su[(size_t)SND[m]*64 + c];
  }
  for (int idx = lane; idx < 16*192; idx += 32) {
    int m = idx / 192, c = idx - m*192;
    VS[m*VSTR + c] = vu[(size_t)SND[m]*192 + c];
  }

  // ---- layer 1: (16x164) @ (164x64 packed), tile-outer, X -> H ----
  #pragma unroll
  for (int t = 0; t < 4; ++t) {
    int col = t*16 + lo;
    v8f acc = mlp_tile<164, 64, XSTR>(X, W1p, col, lo, hi);
    float bb = b1[col];
    #pragma unroll
    for (int r = 0; r < 8; ++r)
      H[(r + 8*hi)*HSTR + col] = silu_f(acc[r] + bb);
  }

  // ---- layer 2: (16x64) @ (64x64 packed), H -> X[:,0:64] ----
  #pragma unroll
  for (int t = 0; t < 4; ++t) {
    int col = t*16 + lo;
    v8f acc = mlp_tile<64, 64, HSTR>(H, W2p, col, lo, hi);
    float bb = b2[col];
    #pragma unroll
    for (int r = 0; r < 8; ++r)
      X[(r + 8*hi)*XSTR + col] = silu_f(acc[r] + bb);
  }

  // ---- layer 3: 5 groups of 64 cols (w0..w4); each tile is consumed
  //      immediately by its tensor-product term + atomic scatter ----
  const float INV_SQRT3 = 0.57735026918962576f;
  const float INV_SQRT2 = 0.70710678118654752f;
  for (int j = 0; j < 5; ++j) {
    #pragma unroll
    for (int t = 0; t < 4; ++t) {
      int f = t*16 + lo;
      v8f acc = mlp_tile<64, 320, XSTR>(X, W3p, j*64 + f, lo, hi);
      #pragma unroll
      for (int r = 0; r < 8; ++r) {
        int m = r + 8*hi;              // edge row within the tile
        float w  = acc[r];             // w_j[edge m, feature f]
        int rcv  = RCV[m];
        float a0  = AT[m*4+0];
        float a1x = AT[m*4+1], a1y = AT[m*4+2], a1z = AT[m*4+3];
        if (j == 0) {                   // p00 * w0 -> M0[:, 0:64]
          atomicAdd(&M0[(size_t)rcv*128 + f], w * SS[m*SSTR + f] * a0);
        } else if (j == 1) {            // p11_0 * w1 -> M0[:, 64:128]
          float vx = VS[m*VSTR + f*3+0], vy = VS[m*VSTR + f*3+1], vz = VS[m*VSTR + f*3+2];
          float p = (vx*a1x + vy*a1y + vz*a1z) * INV_SQRT3;
          atomicAdd(&M0[(size_t)rcv*128 + 64 + f], w * p);
        } else if (j == 2) {            // p01 * w2 -> M1[:, 0:64, :]
          float s = w * SS[m*SSTR + f];
          float* dst = &M1[((size_t)rcv*192 + f)*3];
          atomicAdd(dst+0, s*a1x);
          atomicAdd(dst+1, s*a1y);
          atomicAdd(dst+2, s*a1z);
        } else if (j == 3) {            // p10 * w3 -> M1[:, 64:128, :]
          float sc = w * a0;
          float vx = VS[m*VSTR + f*3+0], vy = VS[m*VSTR + f*3+1], vz = VS[m*VSTR + f*3+2];
          float* dst = &M1[((size_t)rcv*192 + 64 + f)*3];
          atomicAdd(dst+0, vx*sc);
          atomicAdd(dst+1, vy*sc);
          atomicAdd(dst+2, vz*sc);
        } else {                        // cross(vs,a1)*inv_sqrt2 * w4 -> M1[:, 128:192, :]
          float vx = VS[m*VSTR + f*3+0], vy = VS[m*VSTR + f*3+1], vz = VS[m*VSTR + f*3+2];
          float cx = vy*a1z - vz*a1y;
          float cy = vz*a1x - vx*a1z;
          float cz = vx*a1y - vy*a1x;
          float sc = w * INV_SQRT2;
          float* dst = &M1[((size_t)rcv*192 + 128 + f)*3];
          atomicAdd(dst+0, cx*sc);
          atomicAdd(dst+1, cy*sc);
          atomicAdd(dst+2, cz*sc);
        }
      }
    }
  }
}

// ---------------------------------------------------------------------------
// Kernel 3: out0 = M0 @ Wl0 / 32 ; out1[:,:,c] = M1[:,:,c] @ Wl1 / 32
// interleave to (N, 64, 4)
// ---------------------------------------------------------------------------
__global__ __launch_bounds__(256) void output_gemm(
    const float* __restrict__ M0, const float* __restrict__ M1,
    const float* __restrict__ Wl0, const float* __restrict__ Wl1,
    float* __restrict__ out)
{
  int t = blockIdx.x * blockDim.x + threadIdx.x;
  int n = t >> 6;
  int g = t & 63;
  const float* m0 = M0 + (size_t)n * 128;
  float o0 = 0.f;
  for (int k = 0; k < 128; ++k) o0 += m0[k] * Wl0[k*64 + g];
  const float* m1 = M1 + (size_t)n * 576;
  float o1x = 0.f, o1y = 0.f, o1z = 0.f;
  for (int k = 0; k < 192; ++k) {
    float w = Wl1[k*64 + g];
    o1x += m1[k*3+0] * w;
    o1y += m1[k*3+1] * w;
    o1z += m1[k*3+2] * w;
  }
  const float inv = 1.0f / 32.0f;   // / AVG_NEIGH
  size_t o = ((size_t)n*64 + g) * 4;
  out[o+0] = o0  * inv;
  out[o+1] = o1x * inv;
  out[o+2] = o1y * inv;
  out[o+3] = o1z * inv;
}

extern "C" void kernel_launch(void* const* d_in, const int* in_sizes, int n_in,
                              void* d_out, int out_size, void* d_ws, size_t ws_size,
                              hipStream_t stream) {
  const float* node_feats = (const float*)d_in[0];
  const float* edge_attrs = (const float*)d_in[1];
  const float* edge_feats = (const float*)d_in[2];
  const float* lengths    = (const float*)d_in[3];
  const int*   edge_index = (const int*)d_in[4];
  const float* Wsc   = (const float*)d_in[5];
  const float* Wup_s = (const float*)d_in[6];
  const float* Wup_v = (const float*)d_in[7];
  const float* W1 = (const float*)d_in[8];
  const float* b1 = (const float*)d_in[9];
  const float* W2 = (const float*)d_in[10];
  const float* b2 = (const float*)d_in[11];
  const float* W3 = (const float*)d_in[12];
  const float* Wl0 = (const float*)d_in[13];
  const float* Wl1 = (const float*)d_in[14];

  const int N = in_sizes[0] / 256;   // 8192
  const int E = in_sizes[4] / 2;     // 262144

  float* ws = (float*)d_ws;
  float* ns  = ws;                          // N*64
  float* su  = ns  + (size_t)N * 64;        // N*64
  float* vu  = su  + (size_t)N * 64;        // N*192
  float* M0  = vu  + (size_t)N * 192;       // N*128
  float* M1  = M0  + (size_t)N * 128;       // N*576
  float* W1p = M1  + (size_t)N * 576;       // 164*64  pair-packed, zero-padded
  float* W2p = W1p + (size_t)164 * 64;      // 64*64   pair-packed
  float* W3p = W2p + (size_t)64 * 64;       // 64*320  pair-packed

  // zero the scatter accumulators (M0 and M1 are contiguous)
  hipMemsetAsync(M0, 0, (size_t)N * (128 + 576) * sizeof(float), stream);

  pack_pairs<<<(164*64 + 255)/256, 256, 0, stream>>>(W1, W1p, 161, 64, 164*64);
  pack_pairs<<<(64*64  + 255)/256, 256, 0, stream>>>(W2, W2p, 64, 64, 64*64);
  pack_pairs<<<(64*320 + 255)/256, 256, 0, stream>>>(W3, W3p, 64, 320, 64*320);

  node_precompute<<<(N * 64) / 256, 256, 0, stream>>>(
      node_feats, Wsc, Wup_s, Wup_v, ns, su, vu);

  edge_mlp_tp<<<E / 32, 64, 0, stream>>>(
      ns, su, vu, edge_attrs, edge_feats, lengths, edge_index,
      W1p, b1, W2p, b2, W3p, M0, M1, E);

  output_gemm<<<(N * 64) / 256, 256, 0, stream>>>(M0, M1, Wl0, Wl1, (float*)d_out);
}